// EmbeddingLoss_82609400971533
// MI455X (gfx1250) — compile-verified
//
#include <hip/hip_runtime.h>
#include <hip/hip_bf16.h>
#include <math.h>

// ---------------------------------------------------------------------------
// EmbeddingLoss for MI455X (gfx1250, wave32, WMMA).
//
// Heavy-lift: Gram matrix P*P^T via V_WMMA_F32_16X16X4_F32 -> 64MB distance
// matrix kept in workspace (fits in the 192MB global L2), then sequential
// Prim / bitonic sort / union-merge as single-workgroup wave32 kernels.
// ---------------------------------------------------------------------------

#define NPTS   4096
#define KLBL   16
#define NE     4095       // MST edges
#define EPAD   4096

typedef __attribute__((ext_vector_type(2))) float v2f;
typedef __attribute__((ext_vector_type(8))) float v8f;

// output layout (flat float, reference tuple order)
#define O_LOSS 0
#define O_EMST 1
#define O_EU   (1 + 3 * NE)            // 12286
#define O_EV   (O_EU + NE)             // 16381
#define O_D    (O_EV + NE)             // 20476
#define O_RP   (O_D + NE)              // 24571
#define O_RN   (O_RP + NE)             // 28666
#define O_NL   (O_RN + NE)             // 32761

// ---------------------------------------------------------------------------
// Kernel 1: points [N][8] (3 emb channels, 3 scaled coords, 2 zero pad) + |p|^2
// ---------------------------------------------------------------------------
__global__ void el_build_points(const float* __restrict__ emb,
                                float* __restrict__ pts,
                                float* __restrict__ sq) {
  int i = blockIdx.x * blockDim.x + threadIdx.x;
  if (i >= NPTS) return;
  float p[8];
  p[0] = emb[0 * NPTS + i];
  p[1] = emb[1 * NPTS + i];
  p[2] = emb[2 * NPTS + i];
  int z = i >> 8, y = (i >> 4) & 15, x = i & 15;
  p[3] = 0.01f * (float)z;
  p[4] = 0.01f * (float)y;
  p[5] = 0.01f * (float)x;
  p[6] = 0.0f;
  p[7] = 0.0f;
  float s = 0.0f;
#pragma unroll
  for (int q = 0; q < 6; ++q) s += p[q] * p[q];
#pragma unroll
  for (int q = 0; q < 8; ++q) pts[(size_t)i * 8 + q] = p[q];
  sq[i] = s;
}

// ---------------------------------------------------------------------------
// Kernel 2: dense distance matrix via WMMA f32 16x16x4 (K=8 padded -> 2 ops).
// One wave per 16x16 tile; 256x256 tiles. Uniform control flow (EXEC all 1s).
//
// A (16x4 f32): lanes 0-15 -> M=lane, VGPR0=K0 VGPR1=K1; lanes 16-31 -> K2,K3.
// B (4x16 f32): lanes 0-15 -> N=lane, VGPR0=K0 VGPR1=K1; lanes 16-31 -> K2,K3.
// C/D (16x16 f32): VGPR r, lanes 0-15 -> M=r, lanes 16-31 -> M=r+8; N=lane&15.
// ---------------------------------------------------------------------------
__global__ void el_dist_wmma(const float* __restrict__ P,
                             const float* __restrict__ sq,
                             float* __restrict__ Dm) {
  int wid  = threadIdx.x >> 5;
  int lane = threadIdx.x & 31;
  int tile = blockIdx.x * 8 + wid;      // 65536 tiles total
  int ti = tile >> 8;
  int tj = tile & 255;
  int m  = lane & 15;
  int kb = (lane >> 4) << 1;            // 0 or 2

  const float* pr = P + (size_t)(ti * 16 + m) * 8;
  const float* pc = P + (size_t)(tj * 16 + m) * 8;
  v2f a0, a1, b0, b1;
  a0[0] = pr[kb];     a0[1] = pr[kb + 1];
  a1[0] = pr[kb + 4]; a1[1] = pr[kb + 5];
  b0[0] = pc[kb];     b0[1] = pc[kb + 1];
  b1[0] = pc[kb + 4]; b1[1] = pc[kb + 5];

  v8f acc = {};
  acc = __builtin_amdgcn_wmma_f32_16x16x4_f32(false, a0, false, b0,
                                              (short)0, acc, false, false);
  acc = __builtin_amdgcn_wmma_f32_16x16x4_f32(false, a1, false, b1,
                                              (short)0, acc, false, false);

  int col   = tj * 16 + m;
  float sj  = sq[col];
  int rbase = ti * 16 + ((lane >> 4) << 3);
#pragma unroll
  for (int r = 0; r < 8; ++r) {
    int row  = rbase + r;
    float d2 = sq[row] + sj - 2.0f * acc[r];
    Dm[(size_t)row * NPTS + col] = sqrtf(fmaxf(d2, 1e-12f));
  }
}

// ---------------------------------------------------------------------------
// Kernel 3: Prim MST. Single block, 1024 threads (32 waves). Rows of Dm come
// from the L2-resident 64MB matrix. Argmin = 2-level wave32 shuffle reduce
// with smaller-index tie-break (matches jnp.argmin).
// ---------------------------------------------------------------------------
__global__ void el_prim(const float* __restrict__ Dm,
                        int* __restrict__ eu, int* __restrict__ ev,
                        float* __restrict__ ew) {
  __shared__ float md[NPTS];
  __shared__ int   ms[NPTS];
  __shared__ unsigned char intree[NPTS];
  __shared__ float rv[32];
  __shared__ int   ri[32];
  __shared__ int   s_v;

  int t = threadIdx.x;
  int lane = t & 31, wid = t >> 5;
  int base = t * 4;                      // contiguous 4 nodes per thread

#pragma unroll
  for (int q = 0; q < 4; ++q) {
    int j = base + q;
    md[j] = Dm[j];                       // row 0 seeds min_dist
    ms[j] = 0;
    intree[j] = 0;
  }
  if (t == 0) intree[0] = 1;
  __syncthreads();

  for (int e = 0; e < NE; ++e) {
    float best = INFINITY;
    int bidx = NPTS;
#pragma unroll
    for (int q = 0; q < 4; ++q) {
      int j = base + q;
      if (!intree[j]) {
        float v = md[j];
        if (v < best) { best = v; bidx = j; }
      }
    }
    for (int off = 16; off > 0; off >>= 1) {
      float ov = __shfl_down(best, off, 32);
      int   oi = __shfl_down(bidx, off, 32);
      if (ov < best || (ov == best && oi < bidx)) { best = ov; bidx = oi; }
    }
    if (lane == 0) { rv[wid] = best; ri[wid] = bidx; }
    __syncthreads();
    if (wid == 0) {
      float b2 = rv[lane];
      int   i2 = ri[lane];
      for (int off = 16; off > 0; off >>= 1) {
        float ov = __shfl_down(b2, off, 32);
        int   oi = __shfl_down(i2, off, 32);
        if (ov < b2 || (ov == b2 && oi < i2)) { b2 = ov; i2 = oi; }
      }
      if (lane == 0) {
        s_v = i2;
        intree[i2] = 1;
        eu[e] = ms[i2];
        ev[e] = i2;
        ew[e] = b2;
      }
    }
    __syncthreads();
    int v = s_v;
    const float* row = Dm + (size_t)v * NPTS;
#pragma unroll
    for (int q = 0; q < 4; ++q) {
      int j = base + q;
      float dv = row[j];
      if (dv < md[j]) { md[j] = dv; ms[j] = v; }
    }
    __syncthreads();
  }
}

// ---------------------------------------------------------------------------
// Kernel 4: bitonic argsort of 4096 edge weights in LDS (index tie-break
// gives a deterministic order equivalent to a stable argsort).
// ---------------------------------------------------------------------------
__global__ void el_sort(const int* __restrict__ eu, const int* __restrict__ ev,
                        const float* __restrict__ ew,
                        int* __restrict__ seu, int* __restrict__ sev,
                        float* __restrict__ sew) {
  __shared__ float key[EPAD];
  __shared__ int   idx[EPAD];
  int t = threadIdx.x;
  for (int j = t; j < EPAD; j += 1024) {
    key[j] = (j < NE) ? ew[j] : INFINITY;
    idx[j] = j;
  }
  __syncthreads();
  for (int ksz = 2; ksz <= EPAD; ksz <<= 1) {
    for (int js = ksz >> 1; js > 0; js >>= 1) {
      for (int i = t; i < EPAD; i += 1024) {
        int ixj = i ^ js;
        if (ixj > i) {
          bool up = ((i & ksz) == 0);
          float ki = key[i], kj = key[ixj];
          int   ii = idx[i], ij = idx[ixj];
          bool swp = up ? (kj < ki || (kj == ki && ij < ii))
                        : (kj > ki || (kj == ki && ij > ii));
          if (swp) {
            key[i] = kj; key[ixj] = ki;
            idx[i] = ij; idx[ixj] = ii;
          }
        }
      }
      __syncthreads();
    }
  }
  for (int r = t; r < NE; r += 1024) {
    int o = idx[r];
    sew[r] = key[r];
    seu[r] = eu[o];
    sev[r] = ev[o];
  }
}

// ---------------------------------------------------------------------------
// Kernel 5: union-merge over sorted edges + loss terms + edge outputs.
// comp[] in LDS; counts[N][16] in workspace. Wave 0 does the K=16 histogram
// math with width-16 shuffles; all 1024 threads relabel comp.
// ---------------------------------------------------------------------------
__global__ void el_um(const int* __restrict__ seu, const int* __restrict__ sev,
                      const float* __restrict__ sew,
                      const int* __restrict__ labels,
                      const float* __restrict__ mask,
                      float* __restrict__ counts,
                      float* __restrict__ num_pos, float* __restrict__ num_neg,
                      float* __restrict__ scal, float* __restrict__ out) {
  __shared__ int comp[NPTS];
  __shared__ float red[1024];
  int t = threadIdx.x;

  for (int j = t; j < NPTS; j += 1024) comp[j] = j;
  for (int i = t; i < NPTS * KLBL; i += 1024) {
    int node = i >> 4, k = i & 15;
    counts[i] = (labels[node] == k) ? mask[node] : 0.0f;
  }
  __syncthreads();

  for (int e = 0; e < NE; ++e) {
    int u = seu[e], v = sev[e];
    int cu = comp[u], cv = comp[v];
    __syncthreads();
    if (t < 32) {
      int k = t & 15;
      float a = counts[(size_t)cu * KLBL + k];
      float b = counts[(size_t)cv * KLBL + k];
      float dp = a * b, sa = a, sb = b;
      for (int m = 8; m > 0; m >>= 1) {
        dp += __shfl_xor(dp, m, 16);
        sa += __shfl_xor(sa, m, 16);
        sb += __shfl_xor(sb, m, 16);
      }
      if (t == 0) { num_pos[e] = dp; num_neg[e] = sa * sb - dp; }
      if (t < 16) {
        counts[(size_t)cu * KLBL + k] = a + b;
        counts[(size_t)cv * KLBL + k] = 0.0f;   // cu==cv degenerate -> 0, matches ref
      }
    }
    for (int j = t; j < NPTS; j += 1024) {
      if (comp[j] == cv) comp[j] = cu;
    }
    __syncthreads();
  }

  // totals
  float tp = 0.0f, tn = 0.0f;
  for (int e = t; e < NE; e += 1024) { tp += num_pos[e]; tn += num_neg[e]; }
  red[t] = tp; __syncthreads();
  for (int s = 512; s > 0; s >>= 1) { if (t < s) red[t] += red[t + s]; __syncthreads(); }
  float totp = fmaxf(red[0], 1e-10f);
  __syncthreads();
  red[t] = tn; __syncthreads();
  for (int s = 512; s > 0; s >>= 1) { if (t < s) red[t] += red[t + s]; __syncthreads(); }
  float totn = fmaxf(red[0], 1e-10f);
  __syncthreads();

  // ratios, loss, edge outputs
  float lp = 0.0f;
  for (int e = t; e < NE; e += 1024) {
    float d  = sew[e];
    float rp = num_pos[e] / totp;
    float rn = num_neg[e] / totn;
    float m1 = fmaxf(1.0f - d, 0.0f);           // ALPHA = 1.0
    lp += rp * d * d + rn * m1 * m1;
    float fu = (float)seu[e], fv = (float)sev[e];
    out[O_RP + e] = rp;
    out[O_RN + e] = rn;
    out[O_D + e]  = d;
    out[O_EU + e] = fu;
    out[O_EV + e] = fv;
    out[O_EMST + 3 * e + 0] = fu;
    out[O_EMST + 3 * e + 1] = fv;
    out[O_EMST + 3 * e + 2] = d;
  }
  red[t] = lp; __syncthreads();
  for (int s = 512; s > 0; s >>= 1) { if (t < s) red[t] += red[t + s]; __syncthreads(); }
  if (t == 0) scal[0] = red[0];                 // um_loss
}

// ---------------------------------------------------------------------------
// Kernel 6: neighborhood MSE + final loss.
// ---------------------------------------------------------------------------
__global__ void el_final(const float* __restrict__ nb,
                         const float* __restrict__ nbm,
                         const float* __restrict__ nbt,
                         const float* __restrict__ scal,
                         float* __restrict__ out) {
  __shared__ float red[1024];
  int t = threadIdx.x;
  float s = 0.0f;
  for (int i = t; i < 3 * NPTS; i += 1024) {
    float d = nb[i] * nbm[i] - nbt[i];
    s += d * d;
  }
  red[t] = s; __syncthreads();
  for (int st = 512; st > 0; st >>= 1) { if (t < st) red[t] += red[t + st]; __syncthreads(); }
  if (t == 0) {
    float nl = red[0] / (float)(3 * NPTS);
    out[O_NL]   = nl;
    out[O_LOSS] = nl * scal[0];
  }
}

// ---------------------------------------------------------------------------
// Launch. Workspace budget (floats): points 32768 + sq 4096 + Dm 16777216 +
// edges 6*4096 + counts 65536 + pos/neg 2*4096 + scalars  ~= 68 MB.
// ---------------------------------------------------------------------------
extern "C" void kernel_launch(void* const* d_in, const int* in_sizes, int n_in,
                              void* d_out, int out_size, void* d_ws, size_t ws_size,
                              hipStream_t stream) {
  const float* emb    = (const float*)d_in[0];   // [1,3,16,16,16]
  const int*   target = (const int*)d_in[1];     // [1,16,16,16]
  const float* mask   = (const float*)d_in[2];   // [1,16,16,16]
  const float* nb     = (const float*)d_in[3];   // [1,3,16,16,16]
  const float* nbm    = (const float*)d_in[4];
  const float* nbt    = (const float*)d_in[5];
  float* out = (float*)d_out;

  float* ws      = (float*)d_ws;
  float* pts     = ws;                              // 4096*8
  float* sq      = pts + (size_t)NPTS * 8;          // 4096
  float* Dm      = sq + NPTS;                       // 4096*4096
  float* ew      = Dm + (size_t)NPTS * NPTS;        // 4096
  int*   eu      = (int*)(ew + EPAD);               // 4096
  int*   ev      = eu + EPAD;                       // 4096
  float* sew     = (float*)(ev + EPAD);             // 4096
  int*   seu     = (int*)(sew + EPAD);              // 4096
  int*   sev     = seu + EPAD;                      // 4096
  float* counts  = (float*)(sev + EPAD);            // 4096*16
  float* num_pos = counts + (size_t)NPTS * KLBL;    // 4096
  float* num_neg = num_pos + EPAD;                  // 4096
  float* scal    = num_neg + EPAD;                  // 4

  el_build_points<<<NPTS / 256, 256, 0, stream>>>(emb, pts, sq);
  el_dist_wmma<<<(256 * 256) / 8, 256, 0, stream>>>(pts, sq, Dm);
  el_prim<<<1, 1024, 0, stream>>>(Dm, eu, ev, ew);
  el_sort<<<1, 1024, 0, stream>>>(eu, ev, ew, seu, sev, sew);
  el_um<<<1, 1024, 0, stream>>>(seu, sev, sew, target, mask,
                                counts, num_pos, num_neg, scal, out);
  el_final<<<1, 1024, 0, stream>>>(nb, nbm, nbt, scal, out);
}